// YOLOLayer_78022375899238
// MI455X (gfx1250) — compile-verified
//
#include <hip/hip_runtime.h>
#include <hip/hip_bf16.h>

// YOLO head: x[16,255,64,64] f32 -> (boxes[16,3,4096,4], conf[16,3,4096], cls[16,3,4096,80])
// Memory-bound elementwise + channel-major -> channel-last transpose.
// Reads coalesced by construction; cls writes coalesced via LDS staging and
// gfx1250 async LDS->global b128 stores (ASYNCcnt-tracked DMA).

#define NUM_A 3
#define NUM_C 80
#define CH 85            // 5 + NUM_C
#define G 64
#define SPATIAL 4096     // 64*64
#define PLANES 48        // 16 * 3
#define TILE 128
#define THREADS 128
#define LDS_STRIDE 84    // 80 rounded to x4 (16B-aligned rows), 2-way bank conflicts max

#define OUT_BOXES_ELEMS (PLANES * SPATIAL * 4)               // 786432
#define OUT_CONF_OFF    OUT_BOXES_ELEMS                      // 786432
#define OUT_CLS_OFF     (OUT_BOXES_ELEMS + PLANES * SPATIAL) // 983040

#if defined(__gfx1250__) && __has_builtin(__builtin_amdgcn_global_store_async_from_lds_b128)
#define HAVE_ASYNC_LDS_STORE 1
#else
#define HAVE_ASYNC_LDS_STORE 0
#endif

// The builtin's params are pointers to 16-byte int vectors (per hipcc diagnostic),
// in global (AS1) / LDS (AS3) address spaces.
typedef int v4i __attribute__((ext_vector_type(4)));
typedef __attribute__((address_space(1))) v4i* g_v4i_ptr;
typedef __attribute__((address_space(3))) v4i* l_v4i_ptr;

__device__ __forceinline__ float fast_exp(float x) {
  // exp(x) = exp2(x * log2(e)) -> v_exp_f32 (TRANS, co-executes with VALU)
  return __builtin_amdgcn_exp2f(x * 1.44269504088896340736f);
}
__device__ __forceinline__ float fast_sigmoid(float x) {
  return __builtin_amdgcn_rcpf(1.0f + fast_exp(-x));
}

__global__ __launch_bounds__(THREADS)
void yolo_head_kernel(const float* __restrict__ x,
                      const float* __restrict__ anchors,
                      float* __restrict__ out) {
  __shared__ float lds_cls[TILE * LDS_STRIDE]; // 43008 B

  const int plane = blockIdx.x >> 5;           // / (SPATIAL/TILE) == /32 ; 0..47 = b*3 + a
  const int tile  = blockIdx.x & 31;
  const int a     = plane % NUM_A;
  const int t     = threadIdx.x;               // 0..127 -> spatial position in tile
  const int s     = tile * TILE + t;
  const int y0    = s >> 6;                    // grid_x (varies along axis 2)
  const int y1    = s & 63;                    // grid_y (varies along axis 3)

  // Input plane base: x[((b*3+a)*85 + ch)*4096 + s]
  const float* __restrict__ base =
      x + (size_t)plane * (CH * SPATIAL) + (size_t)tile * TILE;

  // ---- box / conf channels (coalesced b32 loads: 32 lanes x 4B) ----
  const float p0 = base[0 * SPATIAL + t];
  const float p1 = base[1 * SPATIAL + t];
  const float p2 = base[2 * SPATIAL + t];
  const float p3 = base[3 * SPATIAL + t];
  const float p4 = base[4 * SPATIAL + t];

  const float aw = anchors[2 * a + 0];  // (anchors[a][0]*g0)/g0 == anchors[a][0]
  const float ah = anchors[2 * a + 1];

  float4 box;
  box.x = (fast_sigmoid(p0) + (float)y0) * 0.015625f; // /64
  box.y = (fast_sigmoid(p1) + (float)y1) * 0.015625f; // /64
  box.z = fast_exp(p2) * aw;
  box.w = fast_exp(p3) * ah;

  // boxes: per-lane b128, lanes contiguous -> fully coalesced
  ((float4*)out)[(size_t)plane * SPATIAL + s] = box;

  // conf: coalesced b32
  out[OUT_CONF_OFF + (size_t)plane * SPATIAL + s] = fast_sigmoid(p4);

  // ---- 80 class channels: coalesced global reads, transpose via LDS ----
  #pragma unroll 16
  for (int c = 0; c < NUM_C; ++c) {
    const float v = fast_sigmoid(base[(5 + c) * SPATIAL + t]);
    lds_cls[t * LDS_STRIDE + c] = v;  // per-lane scatter: 2-way bank conflict max
  }
  __syncthreads();

  // Drain LDS -> global as contiguous b128 chunks.
  // Flat cls floats for this tile = TILE*80 = 10240 -> 2560 float4 chunks, 20/thread.
  float* __restrict__ out_cls =
      out + OUT_CLS_OFF + ((size_t)plane * SPATIAL + (size_t)tile * TILE) * NUM_C;

  #pragma unroll
  for (int k = 0; k < (TILE * NUM_C / 4) / THREADS; ++k) {
    const int q   = k * THREADS + t;  // chunk index; consecutive lanes -> consecutive 16B
    const int pos = q / (NUM_C / 4);  // /20
    const int c4  = q % (NUM_C / 4);
    float* src = &lds_cls[pos * LDS_STRIDE + c4 * 4]; // 16B aligned (stride 84)
#if HAVE_ASYNC_LDS_STORE
    // gfx1250 async DMA: LDS -> global, no VGPR data movement, tracked by ASYNCcnt
    __builtin_amdgcn_global_store_async_from_lds_b128(
        (g_v4i_ptr)(out_cls + 4 * (size_t)q), (l_v4i_ptr)src, 0, 0);
#else
    ((float4*)out_cls)[q] = *(const float4*)src;
#endif
  }

#if HAVE_ASYNC_LDS_STORE && __has_builtin(__builtin_amdgcn_s_wait_asynccnt)
  __builtin_amdgcn_s_wait_asynccnt(0);
#endif
}

extern "C" void kernel_launch(void* const* d_in, const int* in_sizes, int n_in,
                              void* d_out, int out_size, void* d_ws, size_t ws_size,
                              hipStream_t stream) {
  (void)in_sizes; (void)n_in; (void)d_ws; (void)ws_size; (void)out_size;
  const float* x       = (const float*)d_in[0];
  const float* anchors = (const float*)d_in[1];
  float* out           = (float*)d_out;

  // 48 planes * 32 tiles = 1536 blocks of 128 threads (4 wave32 waves each)
  yolo_head_kernel<<<dim3(PLANES * (SPATIAL / TILE)), dim3(THREADS), 0, stream>>>(
      x, anchors, out);
}